// MultiBoxLoss_13615046328876
// MI455X (gfx1250) — compile-verified
//
#include <hip/hip_runtime.h>
#include <math.h>

// Problem constants (from reference)
#define BB 32
#define PP 24564
#define OO 50
#define CC 81
#define THRESH 0.5f

__device__ __forceinline__ float sl1f(float d) {
    float a = fabsf(d);
    return (a < 1.0f) ? 0.5f * d * d : a - 0.5f;
}

// ---------------------------------------------------------------- init
__global__ void mbl_init(unsigned long long* __restrict__ keys,
                         int* __restrict__ num_pos,
                         float* __restrict__ accs) {
    int i = blockIdx.x * blockDim.x + threadIdx.x;
    if (i < BB * OO) keys[i] = 0ull;
    if (i < BB) num_pos[i] = 0;
    if (i < 2) accs[i] = 0.0f;
}

// ---------------------------------------------------------------- match
// grid (ceil(P/256), B). Per prior: max/argmax IoU over 50 truths.
// Per truth: argmax over priors via packed (iou_bits<<32 | ~p) atomicMax.
// Truths staged into LDS with gfx1250 async global->LDS loads (ASYNCcnt).
__global__ void mbl_match(const float* __restrict__ priors,
                          const float* __restrict__ targets,
                          float* __restrict__ bo,
                          int* __restrict__ bi,
                          unsigned long long* __restrict__ keys) {
    __shared__ float s_tr[OO * 4];
    const int b = blockIdx.y;
    const int t = threadIdx.x;

    if (t < OO * 4) {
        const int o = t >> 2, k = t & 3;
        const float* gp = targets + ((size_t)b * OO + o) * 5 + k;
        unsigned lds_off = (unsigned)(size_t)(&s_tr[t]);
        asm volatile("global_load_async_to_lds_b32 %0, %1, off"
                     :: "v"(lds_off), "v"(gp) : "memory");
    }
    asm volatile("s_wait_asynccnt 0x0" ::: "memory");
    __syncthreads();

    const int p = blockIdx.x * blockDim.x + t;
    const bool valid = (p < PP);

    float pl = 0.f, ptp = 0.f, prr = 0.f, pbt = 0.f, area_p = 1.f;
    if (valid) {
        float4 pr = reinterpret_cast<const float4*>(priors)[p];
        pl  = pr.x - pr.z * 0.5f;  ptp = pr.y - pr.w * 0.5f;
        prr = pr.x + pr.z * 0.5f;  pbt = pr.y + pr.w * 0.5f;
        area_p = pr.z * pr.w;
    }

    float best = -1.0f; int bestj = 0;
    for (int j = 0; j < OO; ++j) {
        float iou = 0.0f;
        if (valid) {
            float tl = s_tr[j*4+0], tt = s_tr[j*4+1];
            float tr = s_tr[j*4+2], tb = s_tr[j*4+3];
            float ix = fminf(prr, tr) - fmaxf(pl, tl);
            float iy = fminf(pbt, tb) - fmaxf(ptp, tt);
            ix = fmaxf(ix, 0.0f); iy = fmaxf(iy, 0.0f);
            float inter = ix * iy;
            float at = (tr - tl) * (tb - tt);
            iou = inter / (at + area_p - inter);
            if (iou > best) { best = iou; bestj = j; }
        }
        // wave32 argmax over priors for this truth
        unsigned long long key = valid
            ? ((unsigned long long)__float_as_uint(iou) << 32) |
              (unsigned long long)(~(unsigned)p)
            : 0ull;
        for (int o2 = 16; o2 > 0; o2 >>= 1) {
            unsigned long long other = __shfl_down(key, o2, 32);
            if (other > key) key = other;
        }
        if ((t & 31) == 0)
            atomicMax(&keys[(size_t)b * OO + j], key);
    }

    if (valid) {
        size_t bp = (size_t)b * PP + p;
        bo[bp] = best;
        bi[bp] = bestj;
    }
}

// ---------------------------------------------------------------- fixup
__global__ void mbl_fixup(const unsigned long long* __restrict__ keys,
                          float* __restrict__ bo, int* __restrict__ bi) {
    int b = threadIdx.x;
    if (b >= BB) return;
    for (int j = 0; j < OO; ++j) {            // sequential: later j wins
        unsigned long long key = keys[(size_t)b * OO + j];
        unsigned p = ~(unsigned)(key & 0xFFFFFFFFull);
        size_t bp = (size_t)b * PP + p;
        bo[bp] = 2.0f;
        bi[bp] = j;
    }
}

// ---------------------------------------------------------------- main fused
// grid (ceil(P/256), B). Encode + SmoothL1 + LSE + gather + pos counts.
__global__ void mbl_main(const float* __restrict__ loc_data,
                         const float* __restrict__ conf_data,
                         const float* __restrict__ priors,
                         const float* __restrict__ targets,
                         const float* __restrict__ bo,
                         const int* __restrict__ bi,
                         float* __restrict__ lcm,
                         float* __restrict__ accs,   // [0]=loss_l, [1]=loss_c
                         int* __restrict__ num_pos) {
    const int b = blockIdx.y;
    const int p = blockIdx.x * blockDim.x + threadIdx.x;

    float ll = 0.0f, lc = 0.0f;
    int np = 0;

    if (p < PP) {
        const size_t bp = (size_t)b * PP + p;
        const float* row = conf_data + bp * CC;
        // prefetch the 324B class-score row while we do the encode math
        __builtin_prefetch(row, 0, 0);
        __builtin_prefetch((const char*)row + 128, 0, 0);
        __builtin_prefetch((const char*)row + 256, 0, 0);

        const float ov = bo[bp];
        const int idx = bi[bp];
        const float* tg = targets + ((size_t)b * OO + idx) * 5;
        const int conf = (ov < THRESH) ? 0 : ((int)tg[4] + 1);
        const bool pos = (conf > 0);

        if (pos) {
            float4 pr = reinterpret_cast<const float4*>(priors)[p];
            float tl = tg[0], tt = tg[1], tr = tg[2], tb = tg[3];
            float gcx = ((tl + tr) * 0.5f - pr.x) / (0.1f * pr.z);
            float gcy = ((tt + tb) * 0.5f - pr.y) / (0.1f * pr.w);
            float gw  = __logf((tr - tl) / pr.z) * 5.0f;   // /0.2
            float gh  = __logf((tb - tt) / pr.w) * 5.0f;
            float4 l = reinterpret_cast<const float4*>(loc_data)[bp];
            ll = sl1f(l.x - gcx) + sl1f(l.y - gcy) +
                 sl1f(l.z - gw)  + sl1f(l.w - gh);
            np = 1;
        }

        // ---- pass 1: max + gather (row is 4B-aligned; peel to 16B, b128 body)
        float m = -INFINITY, g = 0.0f;
        const int mis = (int)(((size_t)row >> 2) & 3);
        const int pre = (4 - mis) & 3;
        for (int i = 0; i < pre; ++i) {
            float v = row[i]; m = fmaxf(m, v); if (i == conf) g = v;
        }
        const int nv = (CC - pre) >> 2;
        const float4* r4 = reinterpret_cast<const float4*>(row + pre);
        for (int q = 0; q < nv; ++q) {
            float4 v = r4[q];
            int base = pre + q * 4;
            m = fmaxf(fmaxf(m, fmaxf(v.x, v.y)), fmaxf(v.z, v.w));
            if ((unsigned)(conf - base) < 4u)
                g = (conf == base) ? v.x : (conf == base+1) ? v.y
                  : (conf == base+2) ? v.z : v.w;
        }
        for (int i = pre + nv * 4; i < CC; ++i) {
            float v = row[i]; m = fmaxf(m, v); if (i == conf) g = v;
        }
        // ---- pass 2: sum exp(v - m)  (row hot in WGP$/L2)
        float s = 0.0f;
        for (int i = 0; i < pre; ++i) s += __expf(row[i] - m);
        for (int q = 0; q < nv; ++q) {
            float4 v = r4[q];
            s += __expf(v.x - m) + __expf(v.y - m) +
                 __expf(v.z - m) + __expf(v.w - m);
        }
        for (int i = pre + nv * 4; i < CC; ++i) s += __expf(row[i] - m);

        const float lca = m + __logf(s) - g;     // lse - gathered  (>= 0)
        lc = pos ? lca : 0.0f;
        lcm[bp] = pos ? 0.0f : lca;
    }

    // wave32 reductions, one atomic per wave
    for (int o2 = 16; o2 > 0; o2 >>= 1) {
        ll += __shfl_down(ll, o2, 32);
        lc += __shfl_down(lc, o2, 32);
        np += __shfl_down(np, o2, 32);
    }
    if ((threadIdx.x & 31) == 0) {
        if (ll != 0.0f) atomicAdd(&accs[0], ll);
        if (lc != 0.0f) atomicAdd(&accs[1], lc);
        if (np != 0)    atomicAdd(&num_pos[b], np);
    }
}

// ---------------------------------------------------------------- top-k sum
// One block per batch. 4-pass radix select on uint bits (all values >= 0),
// then sum of strict-greater values + krem * threshold.
__global__ void mbl_topk(const float* __restrict__ lcm,
                         const int* __restrict__ num_pos,
                         float* __restrict__ accs) {
    __shared__ unsigned hist[256];
    __shared__ unsigned s_prefix, s_krem;
    __shared__ float warpsum[8];

    const int b = blockIdx.x;
    const int t = threadIdx.x;

    long long kk = 3LL * (long long)num_pos[b];
    if (kk > PP - 1) kk = PP - 1;
    if (kk <= 0) return;                  // uniform across block

    const float* vals = lcm + (size_t)b * PP;
    if (t == 0) { s_prefix = 0u; s_krem = (unsigned)kk; }

    unsigned prefix_mask = 0u;
    for (int pass = 0; pass < 4; ++pass) {
        const int shift = 24 - 8 * pass;
        hist[t] = 0u;
        __syncthreads();
        const unsigned prefix = s_prefix;
        for (int i = t; i < PP; i += 256) {
            unsigned ub = __float_as_uint(vals[i]);
            if ((ub & prefix_mask) == prefix)
                atomicAdd(&hist[(ub >> shift) & 0xFFu], 1u);
        }
        __syncthreads();
        if (t == 0) {
            unsigned krem = s_krem, cum = 0u;
            int v = 255;
            for (; v > 0; --v) {
                if (cum + hist[v] >= krem) break;
                cum += hist[v];
            }
            s_krem = krem - cum;
            s_prefix = prefix | ((unsigned)v << shift);
        }
        __syncthreads();
        prefix_mask |= 0xFFu << shift;
    }

    const unsigned tbits = s_prefix;
    const unsigned krem  = s_krem;
    float sum = 0.0f;
    for (int i = t; i < PP; i += 256) {
        float v = vals[i];
        if (__float_as_uint(v) > tbits) sum += v;
    }
    for (int o2 = 16; o2 > 0; o2 >>= 1) sum += __shfl_down(sum, o2, 32);
    if ((t & 31) == 0) warpsum[t >> 5] = sum;
    __syncthreads();
    if (t == 0) {
        float tot = 0.0f;
        for (int w = 0; w < 8; ++w) tot += warpsum[w];
        tot += (float)krem * __uint_as_float(tbits);
        atomicAdd(&accs[1], tot);
    }
}

// ---------------------------------------------------------------- finalize
__global__ void mbl_final(const float* __restrict__ accs,
                          const int* __restrict__ num_pos,
                          float* __restrict__ out) {
    if (threadIdx.x == 0 && blockIdx.x == 0) {
        int N = 0;
        for (int b = 0; b < BB; ++b) N += num_pos[b];
        float fN = (float)N;
        out[0] = accs[0] / fN;
        out[1] = accs[1] / fN;
    }
}

// ---------------------------------------------------------------- launch
extern "C" void kernel_launch(void* const* d_in, const int* in_sizes, int n_in,
                              void* d_out, int out_size, void* d_ws, size_t ws_size,
                              hipStream_t stream) {
    const float* loc_data  = (const float*)d_in[0];   // B*P*4
    const float* conf_data = (const float*)d_in[1];   // B*P*81
    const float* priors    = (const float*)d_in[2];   // P*4
    const float* targets   = (const float*)d_in[3];   // B*O*5
    float* out = (float*)d_out;

    const size_t S = (size_t)BB * PP;                 // 786048
    char* ws = (char*)d_ws;
    float*              bo      = (float*)(ws);
    int*                bi      = (int*)  (ws + S * 4);
    float*              lcm     = (float*)(ws + S * 8);
    unsigned long long* keys    = (unsigned long long*)(ws + S * 12);
    int*                num_pos = (int*)  (ws + S * 12 + (size_t)BB * OO * 8);
    float*              accs    = (float*)(ws + S * 12 + (size_t)BB * OO * 8 + 128);

    const int PB = (PP + 255) / 256;                  // 96

    mbl_init <<<dim3((BB * OO + 255) / 256), dim3(256), 0, stream>>>(keys, num_pos, accs);
    mbl_match<<<dim3(PB, BB), dim3(256), 0, stream>>>(priors, targets, bo, bi, keys);
    mbl_fixup<<<dim3(1), dim3(32), 0, stream>>>(keys, bo, bi);
    mbl_main <<<dim3(PB, BB), dim3(256), 0, stream>>>(loc_data, conf_data, priors, targets,
                                                      bo, bi, lcm, accs, num_pos);
    mbl_topk <<<dim3(BB), dim3(256), 0, stream>>>(lcm, num_pos, accs);
    mbl_final<<<dim3(1), dim3(32), 0, stream>>>(accs, num_pos, out);
}